// DPLayer_45784351375496
// MI455X (gfx1250) — compile-verified
//
#include <hip/hip_runtime.h>
#include <math.h>

// Min-plus anti-diagonal DP over 128x128 images, softplus edge weights.
// One wave32 per image, 4 rows per lane, wave-synchronous (zero barriers).
// Image staged to LDS with CDNA5 async global->LDS DMA (ASYNCcnt).
// Branchless inner body: exec-masked lanes cost the same issue slots as
// predicated lanes, so predication removes all SALU/branch overhead.

#define II 128
#define JJ 128
#define WAVES_PER_BLOCK 4   // 4 waves * 64KB = 256KB static LDS (<320KB/WGP)

__device__ __forceinline__ float softplus_f(float x) {
    // log(1+exp(x)) stable: max(x,0) + log(1+exp(-|x|))
    float e = __expf(-fabsf(x));
    return fmaxf(x, 0.0f) + __logf(1.0f + e);
}

__global__ __launch_bounds__(32 * WAVES_PER_BLOCK, 1)
void dp_wavefront_kernel(const float* __restrict__ images,
                         float* __restrict__ out, int B) {
    __shared__ float lds[WAVES_PER_BLOCK * II * JJ];

    const int lane = threadIdx.x & 31;
    const int wid  = threadIdx.x >> 5;
    const int img  = blockIdx.x * WAVES_PER_BLOCK + wid;
    if (img >= B) return;   // uniform per wave

    // ---- Stage this wave's image into its private LDS slice ----------------
    // global_load_async_to_lds_b128: 32 lanes x 16B = 512B coalesced per
    // issue; 128 issues = 64KB. Tracked with ASYNCcnt.
    const unsigned ldsWaveByteBase = (unsigned)(wid * II * JJ * 4);
    const char* gbase = (const char*)(images + (size_t)img * (II * JJ));
    #pragma unroll 4
    for (int it = 0; it < (II * JJ * 4) / (32 * 16); ++it) {
        unsigned off   = (unsigned)(it * 32 + lane) * 16u;
        unsigned laddr = ldsWaveByteBase + off;
        const char* g  = gbase + off;
        asm volatile("global_load_async_to_lds_b128 %0, %1, off"
                     :: "v"(laddr), "v"(g) : "memory");
    }
    asm volatile("s_wait_asynccnt 0" ::: "memory");

    const float* __restrict__ x = &lds[wid * II * JJ];

    // ---- DP state: lane owns rows r = 4*lane + q, q = 0..3 -----------------
    const float INF = __builtin_inff();
    float v[4]  = {INF, INF, INF, INF};   // v[q] = DP value of row (4*lane+q)
    float xp[4] = {0.f, 0.f, 0.f, 0.f};  // xp[q] = last x[r][j] loaded
    if (lane == 0) {
        v[0]  = 0.0f;     // cell (0,0) at d=0
        xp[0] = x[0];     // x[0][0]
    }
    const int rbase = lane * 4;

    // ---- Anti-diagonal sweep d = 1..254 ------------------------------------
    #pragma unroll 2
    for (int d = 1; d <= II + JJ - 2; ++d) {
        // Old values from up-neighbor lane's last row (captured pre-update)
        float xup_in = __shfl_up(xp[3], 1, 32);
        float vup_in = __shfl_up(v[3], 1, 32);

        // Phase A: LDS loads (clamped index -> always in-bounds, branchless)
        float xc[4];
        bool  ok[4];
        #pragma unroll
        for (int q = 0; q < 4; ++q) {
            int r = rbase + q;
            int j = d - r;
            ok[q] = ((unsigned)j < (unsigned)JJ);
            int jc = j < 0 ? 0 : (j > JJ - 1 ? JJ - 1 : j);
            xc[q] = x[r * JJ + jc];
        }

        // Phase B: edge weights (independent of the v-chain; 8 softplus of ILP)
        float th[4], tvv[4];
        #pragma unroll
        for (int q = 0; q < 4; ++q) {
            int r = rbase + q;
            int j = d - r;
            float xu = (q == 0) ? xup_in : xp[q - 1];   // x[r-1][j]
            th[q]  = (j >= 1) ? softplus_f(0.5f * (xp[q] + xc[q])) : 0.0f;
            tvv[q] = softplus_f(0.5f * (xu + xc[q]));
        }

        // Phase C: v updates, descending q so v[q-1] is read pre-update
        #pragma unroll
        for (int q = 3; q >= 0; --q) {
            int r = rbase + q;
            float vu = (q == 0) ? vup_in : v[q - 1];    // v(r-1, j)
            float fl = v[q] + th[q];                    // from-left
            float fu = (r >= 1) ? (vu + tvv[q]) : INF;  // from-up (row 0: none)
            float vn = fminf(fl, fu);
            v[q] = ok[q] ? vn : v[q];
        }

        // Phase D: commit xp
        #pragma unroll
        for (int q = 0; q < 4; ++q)
            xp[q] = ok[q] ? xc[q] : xp[q];
    }

    // Answer: cell (127,127) lives in lane 31, q=3 after d=254.
    if (lane == 31) out[img] = v[3];
}

extern "C" void kernel_launch(void* const* d_in, const int* in_sizes, int n_in,
                              void* d_out, int out_size, void* d_ws, size_t ws_size,
                              hipStream_t stream) {
    const float* images = (const float*)d_in[0];
    float* out = (float*)d_out;
    int B = in_sizes[0] / (II * JJ);
    int blocks = (B + WAVES_PER_BLOCK - 1) / WAVES_PER_BLOCK;
    hipLaunchKernelGGL(dp_wavefront_kernel, dim3(blocks),
                       dim3(32 * WAVES_PER_BLOCK), 0, stream, images, out, B);
}